// GFlowNetActor_41016937677178
// MI455X (gfx1250) — compile-verified
//
#include <hip/hip_runtime.h>
#include <math.h>

// ---------------------------------------------------------------------------
// GFlowNet actor: per-graph (segment) double log-softmax + greedy action.
// edge_batch is sorted -> each graph owns a contiguous edge range.
// One 256-thread block (8 wave32) per graph. LDS caches the masked base
// logits + residuals so HBM is touched once per edge (+ one output write).
// CDNA5 paths: global_load_async_to_lds_b32 (ASYNCcnt) stages edge_residual
// into LDS overlapped with pass-1/2 compute; global_prefetch_b8 on streams.
// ---------------------------------------------------------------------------

#define LNEG   (-1.0e9f)
#define FEPS_  (1.1920928955078125e-7f)   // np.finfo(float32).eps

constexpr int NT  = 256;        // threads per block
constexpr int NW  = NT / 32;    // wave32 waves per block
constexpr int CAP = 8192;       // LDS-cached edges per graph (64 KB)

__device__ __forceinline__ float blk_max(float v, float* red, float* bc) {
  for (int o = 16; o; o >>= 1) v = fmaxf(v, __shfl_xor(v, o));
  const int lane = threadIdx.x & 31, wave = threadIdx.x >> 5;
  if (lane == 0) red[wave] = v;
  __syncthreads();
  if (threadIdx.x < 32) {
    float m = (threadIdx.x < NW) ? red[threadIdx.x] : -3.402823466e38f;
    for (int o = NW >> 1; o; o >>= 1) m = fmaxf(m, __shfl_xor(m, o));
    if (threadIdx.x == 0) bc[0] = m;
  }
  __syncthreads();
  return bc[0];
}

__device__ __forceinline__ float blk_sum(float v, float* red, float* bc) {
  for (int o = 16; o; o >>= 1) v += __shfl_xor(v, o);
  const int lane = threadIdx.x & 31, wave = threadIdx.x >> 5;
  if (lane == 0) red[wave] = v;
  __syncthreads();
  if (threadIdx.x < 32) {
    float s = (threadIdx.x < NW) ? red[threadIdx.x] : 0.0f;
    for (int o = NW >> 1; o; o >>= 1) s += __shfl_xor(s, o);
    if (threadIdx.x == 0) bc[0] = s;
  }
  __syncthreads();
  return bc[0];
}

__global__ __launch_bounds__(NT)
void gfn_actor_kernel(const float* __restrict__ scores,
                      const float* __restrict__ eres,
                      const float* __restrict__ sres,
                      const int*   __restrict__ valid,
                      const int*   __restrict__ ebatch,
                      float* __restrict__ out_edge,   // [E]
                      float* __restrict__ out_stop,   // [G]
                      float* __restrict__ out_act,    // [G]
                      float* __restrict__ out_pf,     // [G]
                      int E)
{
  __shared__ float s_ml[CAP];     // masked base logit (LNEG marks invalid)
  __shared__ float s_res[CAP];    // edge residual (filled by async-to-LDS)
  __shared__ float redf[NW];
  __shared__ float redf2[NW];
  __shared__ int   redi[NW];
  __shared__ float bc[2];
  __shared__ int   bounds[2];

  const int g   = blockIdx.x;
  const int tid = threadIdx.x;

  // --- segment bounds via binary search on sorted edge_batch ---
  if (tid == 0) {
    int lo = 0, hi = E;
    while (lo < hi) { int mid = (lo + hi) >> 1; if (ebatch[mid] < g) lo = mid + 1; else hi = mid; }
    bounds[0] = lo;
    int lo2 = lo, hi2 = E;
    while (lo2 < hi2) { int mid = (lo2 + hi2) >> 1; if (ebatch[mid] < g + 1) lo2 = mid + 1; else hi2 = mid; }
    bounds[1] = lo2;
  }
  __syncthreads();
  const int start  = bounds[0];
  const int end    = bounds[1];
  const int count  = end - start;
  const bool cached = (count <= CAP);

  // --- Pass 1: base masked logits -> max + valid count (fills LDS cache) ---
  // In the cached path, edge_residual is copied global->LDS with the CDNA5
  // async copy engine (ASYNCcnt); it overlaps pass-1/pass-2 compute and is
  // only consumed from pass 3 onward.
  float lmax = LNEG, lcnt = 0.0f;
  if (cached) {
    const float* res_base = eres + start;          // uniform SGPR base (GVS mode)
    for (int j = tid; j < count; j += NT) {
      const int i = start + j;
      __builtin_prefetch(scores + i + 4 * NT, 0, 0);
      {
        const unsigned lds_off = (unsigned)(uintptr_t)(&s_res[j]); // LDS byte offset
        const unsigned goff    = (unsigned)j * 4u;                  // byte offset vs base
        asm volatile("global_load_async_to_lds_b32 %0, %1, %2"
                     :
                     : "v"(lds_off), "v"(goff), "s"(res_base)
                     : "memory");
      }
      const float sc = scores[i];
      const int   v  = valid[i];
      const float ml = v ? logf(fmaxf(sc, 1.0e-6f)) : LNEG;
      s_ml[j]  = ml;
      lmax = fmaxf(lmax, ml);
      lcnt += v ? 1.0f : 0.0f;
    }
  } else {
    for (int i = start + tid; i < end; i += NT) {
      __builtin_prefetch(scores + i + 4 * NT, 0, 0);
      const float sc = scores[i];
      const int   v  = valid[i];
      const float ml = v ? logf(fmaxf(sc, 1.0e-6f)) : LNEG;
      lmax = fmaxf(lmax, ml);
      lcnt += v ? 1.0f : 0.0f;
    }
  }
  const float max_edge  = blk_max(lmax, redf, bc);   // == LNEG if empty/all-invalid
  const float valid_cnt = blk_sum(lcnt, redf, bc);
  const float mj1 = fmaxf(max_edge, 0.0f);           // stop logit = 0, temp = 1

  // --- Pass 2: base sum-exp ---
  float lsum = 0.0f;
  if (cached) {
    for (int j = tid; j < count; j += NT) lsum += expf(s_ml[j] - mj1);
  } else {
    for (int i = start + tid; i < end; i += NT) {
      const float sc = scores[i];
      const int   v  = valid[i];
      const float ml = v ? logf(fmaxf(sc, 1.0e-6f)) : LNEG;
      lsum += expf(ml - mj1);
    }
  }
  // Drain this wave's async copies; the barrier inside blk_sum makes the
  // staged s_res visible block-wide before pass 3 reads it.
  asm volatile("s_wait_asynccnt 0x0" ::: "memory");
  const float sum1  = blk_sum(lsum, redf, bc);
  const float ld1   = mj1 + logf(sum1 + expf(0.0f - mj1) + FEPS_);
  const float cstop = (0.0f - ld1) + sres[g];        // combined stop logit (temp=1)

  // --- Pass 3: combined-logit max ---
  float lmax2 = LNEG;
  if (cached) {
    for (int j = tid; j < count; j += NT) {
      const float ml = s_ml[j];
      const float m2 = (ml > -1.0e8f) ? (ml - ld1 + s_res[j]) : LNEG;
      lmax2 = fmaxf(lmax2, m2);
    }
  } else {
    for (int i = start + tid; i < end; i += NT) {
      const float sc = scores[i];
      const int   v  = valid[i];
      const float m2 = v ? (logf(fmaxf(sc, 1.0e-6f)) - ld1 + eres[i]) : LNEG;
      lmax2 = fmaxf(lmax2, m2);
    }
  }
  const float max2 = blk_max(lmax2, redf, bc);
  const float mj2  = fmaxf(max2, cstop);

  // --- Pass 4: combined sum-exp ---
  float lsum2 = 0.0f;
  if (cached) {
    for (int j = tid; j < count; j += NT) {
      const float ml = s_ml[j];
      const float m2 = (ml > -1.0e8f) ? (ml - ld1 + s_res[j]) : LNEG;
      lsum2 += expf(m2 - mj2);
    }
  } else {
    for (int i = start + tid; i < end; i += NT) {
      const float sc = scores[i];
      const int   v  = valid[i];
      const float m2 = v ? (logf(fmaxf(sc, 1.0e-6f)) - ld1 + eres[i]) : LNEG;
      lsum2 += expf(m2 - mj2);
    }
  }
  const float sum2 = blk_sum(lsum2, redf, bc);
  const float ld2  = mj2 + logf(sum2 + expf(cstop - mj2) + FEPS_);

  const float cls       = cstop - ld2;               // clean_log_stop
  const float inv_total = 1.0f / (valid_cnt + 1.0f); // max(cnt+1,1) == cnt+1
  const float samp_stop = 0.9f * expf(cls) + 0.1f * inv_total;
  const float lss       = logf(fmaxf(samp_stop, 1.0e-12f));

  // --- Pass 5: write clean_log_edge + argmax of log_sample_edge ---
  float bval  = -INFINITY;
  int   bidx  = E;
  float bclean = LNEG;
  if (cached) {
    for (int j = tid; j < count; j += NT) {
      const int   i  = start + j;
      const float ml = s_ml[j];
      const bool  v  = (ml > -1.0e8f);
      const float m2 = v ? (ml - ld1 + s_res[j]) : LNEG;
      const float clean = v ? (m2 - ld2) : LNEG;
      out_edge[i] = clean;
      const float ls = v ? logf(fmaxf(0.9f * expf(clean) + 0.1f * inv_total, 1.0e-12f))
                         : LNEG;
      if (ls > bval || (ls == bval && i < bidx)) { bval = ls; bidx = i; bclean = clean; }
    }
  } else {
    for (int i = start + tid; i < end; i += NT) {
      const float sc = scores[i];
      const bool  v  = (valid[i] != 0);
      const float m2 = v ? (logf(fmaxf(sc, 1.0e-6f)) - ld1 + eres[i]) : LNEG;
      const float clean = v ? (m2 - ld2) : LNEG;
      out_edge[i] = clean;
      const float ls = v ? logf(fmaxf(0.9f * expf(clean) + 0.1f * inv_total, 1.0e-12f))
                         : LNEG;
      if (ls > bval || (ls == bval && i < bidx)) { bval = ls; bidx = i; bclean = clean; }
    }
  }
  // wave32 argmax reduce (max value, min index on ties, carry clean lp)
  for (int o = 16; o; o >>= 1) {
    const float ov = __shfl_xor(bval, o);
    const int   oi = __shfl_xor(bidx, o);
    const float oc = __shfl_xor(bclean, o);
    if (ov > bval || (ov == bval && oi < bidx)) { bval = ov; bidx = oi; bclean = oc; }
  }
  const int lane = tid & 31, wave = tid >> 5;
  if (lane == 0) { redf[wave] = bval; redi[wave] = bidx; redf2[wave] = bclean; }
  __syncthreads();
  if (tid == 0) {
    float v0 = redf[0]; int i0 = redi[0]; float c0 = redf2[0];
    for (int w = 1; w < NW; ++w) {
      if (redf[w] > v0 || (redf[w] == v0 && redi[w] < i0)) {
        v0 = redf[w]; i0 = redi[w]; c0 = redf2[w];
      }
    }
    const bool take_stop = (lss >= v0);
    out_stop[g] = cls;
    out_act[g]  = (float)(take_stop ? end : i0);   // end == cumsum(counts)[g]
    out_pf[g]   = take_stop ? cls : c0;
  }
}

extern "C" void kernel_launch(void* const* d_in, const int* in_sizes, int n_in,
                              void* d_out, int out_size, void* d_ws, size_t ws_size,
                              hipStream_t stream) {
  const float* scores = (const float*)d_in[0];   // edge_scores   [E]
  const float* eres   = (const float*)d_in[1];   // edge_residual [E]
  const float* sres   = (const float*)d_in[2];   // stop_residual [G]
  const int*   valid  = (const int*)d_in[3];     // valid_edges   [E]
  const int*   ebatch = (const int*)d_in[4];     // edge_batch    [E] (sorted)
  const int E = in_sizes[0];
  const int G = in_sizes[2];

  float* out      = (float*)d_out;
  float* out_edge = out;                             // [E]
  float* out_stop = out + (size_t)E;                 // [G]
  float* out_act  = out + (size_t)E + G;             // [G]
  float* out_pf   = out + (size_t)E + 2 * (size_t)G; // [G]

  gfn_actor_kernel<<<G, NT, 0, stream>>>(scores, eres, sres, valid, ebatch,
                                         out_edge, out_stop, out_act, out_pf, E);
}